// GATHK5_8031588843987
// MI455X (gfx1250) — compile-verified
//
#include <hip/hip_runtime.h>
#include <hip/hip_bf16.h>
#include <math.h>

#define N_NODES 100000
#define N_EDGES 3200000

typedef float v2f __attribute__((ext_vector_type(2)));
typedef float v8f __attribute__((ext_vector_type(8)));

// ---- monotone float<->uint mapping for atomic max on f32 via u32 atomics ----
__device__ __forceinline__ unsigned f2ord(float f) {
    unsigned u = __float_as_uint(f);
    return (u & 0x80000000u) ? ~u : (u | 0x80000000u);
}
__device__ __forceinline__ float ord2f(unsigned u) {
    return (u & 0x80000000u) ? __uint_as_float(u & 0x7fffffffu)
                             : __uint_as_float(~u);
}

__device__ __forceinline__ float lrelu02(float v) { return v > 0.f ? v : 0.2f * v; }

// ---------------------------------------------------------------------------
// H[n, NCOLS] = X[n, K] @ W[K, NCOLS]   via V_WMMA_F32_16X16X4_F32
// Compile-time K/NCOLS -> fully unrolled K-loop, no EXEC masking in the body.
// A 16x4 f32 layout: lanes 0-15 hold K=k0,k0+1 ; lanes 16-31 hold K=k0+2,k0+3
// B 4x16 f32 layout: symmetric (N across lanes, K pair per half-wave)
// D 16x16 f32: lane half h -> rows M = v + 8*h, col N = lane&15
// ---------------------------------------------------------------------------
template <int K, int NCOLS>
__global__ void k_proj_t(const float* __restrict__ X, const float* __restrict__ W,
                         float* __restrict__ H, int n) {
    const int lane = threadIdx.x & 31;
    const int tile = blockIdx.x * (blockDim.x >> 5) + (threadIdx.x >> 5);
    if (tile * 16 >= n) return;                 // wave-uniform: EXEC stays full
    const int row0 = tile * 16;
    const int half = lane >> 4;
    const int r    = lane & 15;

    v8f acc0 = {};
    v8f acc1 = {};
    constexpr int NKB = (K + 3) / 4;
#pragma unroll
    for (int kb = 0; kb < NKB; ++kb) {
        const int k0 = kb * 4 + 2 * half;       // even, uniform per half-wave
        v2f a, b0, b1;
        if constexpr ((K % 4) == 0) {
            const float* ap = &X[(row0 + r) * K + k0];
            a.x  = ap[0];
            a.y  = ap[1];
            b0.x = W[ k0      * NCOLS + r];
            b0.y = W[(k0 + 1) * NCOLS + r];
            if constexpr (NCOLS == 32) {
                b1.x = W[ k0      * NCOLS + 16 + r];
                b1.y = W[(k0 + 1) * NCOLS + 16 + r];
            }
        } else {
            // tail guards: unconditional clamped load + select (no branches)
            const bool ok0 = (k0     < K);
            const bool ok1 = (k0 + 1 < K);
            const float a0 = X[(row0 + r) * K + (ok0 ? k0 : 0)];
            const float a1 = X[(row0 + r) * K + (ok1 ? k0 + 1 : 0)];
            a.x  = ok0 ? a0 : 0.f;
            a.y  = ok1 ? a1 : 0.f;
            const float w0 = W[(ok0 ? k0 : 0)     * NCOLS + r];
            const float w1 = W[(ok1 ? k0 + 1 : 0) * NCOLS + r];
            b0.x = ok0 ? w0 : 0.f;
            b0.y = ok1 ? w1 : 0.f;
            if constexpr (NCOLS == 32) {
                const float w2 = W[(ok0 ? k0 : 0)     * NCOLS + 16 + r];
                const float w3 = W[(ok1 ? k0 + 1 : 0) * NCOLS + 16 + r];
                b1.x = ok0 ? w2 : 0.f;
                b1.y = ok1 ? w3 : 0.f;
            }
        }
        acc0 = __builtin_amdgcn_wmma_f32_16x16x4_f32(
                   false, a, false, b0, (short)0, acc0, false, false);
        if constexpr (NCOLS == 32) {
            acc1 = __builtin_amdgcn_wmma_f32_16x16x4_f32(
                       false, a, false, b1, (short)0, acc1, false, false);
        }
    }
#pragma unroll
    for (int v = 0; v < 8; ++v) {
        const int m = v + 8 * half;
        H[(row0 + m) * NCOLS + r] = acc0[v];
        if constexpr (NCOLS == 32) H[(row0 + m) * NCOLS + 16 + r] = acc1[v];
    }
}

// per-node attention logits + init of m / s / agg accumulators
__global__ void k_alpha(const float* __restrict__ H,
                        const float* __restrict__ a_src, const float* __restrict__ a_dst,
                        float* __restrict__ asrc, float* __restrict__ adst,
                        unsigned* __restrict__ m_u, float* __restrict__ s,
                        float* __restrict__ agg, int n, int heads, int ncols) {
    const int i = blockIdx.x * blockDim.x + threadIdx.x;
    if (i >= n) return;
    for (int h = 0; h < heads; ++h) {
        float s1 = 0.f, s2 = 0.f;
#pragma unroll
        for (int c = 0; c < 16; ++c) {
            const float v = H[i * ncols + h * 16 + c];
            s1 += v * a_src[h * 16 + c];
            s2 += v * a_dst[h * 16 + c];
        }
        asrc[i * heads + h] = s1;
        adst[i * heads + h] = s2;
        m_u[i * heads + h]  = f2ord(-INFINITY);
        s[i * heads + h]    = 0.f;
    }
    for (int c = 0; c < ncols; ++c) agg[i * ncols + c] = 0.f;
}

__device__ __forceinline__ void edge_sd(const int* __restrict__ ei, int e, int E,
                                        int& src, int& dst) {
    if (e < E) { src = ei[e]; dst = ei[E + e]; }
    else       { src = dst = e - E; }          // self-loops appended
}

__global__ void k_edge_max(const int* __restrict__ ei,
                           const float* __restrict__ asrc, const float* __restrict__ adst,
                           unsigned* __restrict__ m_u, int E, int Etot, int heads) {
    const int e = blockIdx.x * blockDim.x + threadIdx.x;
    if (e >= Etot) return;
    int src, dst; edge_sd(ei, e, E, src, dst);
    for (int h = 0; h < heads; ++h) {
        const float v = lrelu02(asrc[src * heads + h] + adst[dst * heads + h]);
        atomicMax(&m_u[dst * heads + h], f2ord(v));
    }
}

__global__ void k_edge_sum(const int* __restrict__ ei,
                           const float* __restrict__ asrc, const float* __restrict__ adst,
                           const unsigned* __restrict__ m_u, float* __restrict__ s,
                           int E, int Etot, int heads) {
    const int e = blockIdx.x * blockDim.x + threadIdx.x;
    if (e >= Etot) return;
    int src, dst; edge_sd(ei, e, E, src, dst);
    for (int h = 0; h < heads; ++h) {
        const float v = lrelu02(asrc[src * heads + h] + adst[dst * heads + h]);
        float m = ord2f(m_u[dst * heads + h]);
        if (!isfinite(m)) m = 0.f;
        atomicAdd(&s[dst * heads + h], __expf(v - m));
    }
}

// one thread per (edge, head): 16-channel weighted scatter-add
__global__ void k_edge_agg(const int* __restrict__ ei,
                           const float* __restrict__ asrc, const float* __restrict__ adst,
                           const unsigned* __restrict__ m_u, const float* __restrict__ s,
                           const float* __restrict__ H, float* __restrict__ agg,
                           int E, int Etot, int heads, int ncols) {
    const int gid = blockIdx.x * blockDim.x + threadIdx.x;
    if (gid >= Etot * heads) return;
    const int e = gid / heads;
    const int h = gid - e * heads;
    int src, dst; edge_sd(ei, e, E, src, dst);
    const float v = lrelu02(asrc[src * heads + h] + adst[dst * heads + h]);
    float m = ord2f(m_u[dst * heads + h]);
    if (!isfinite(m)) m = 0.f;
    const float alpha = __expf(v - m) / (s[dst * heads + h] + 1e-16f);

    const float4* hs = (const float4*)&H[src * ncols + h * 16];
    float* ad = &agg[dst * ncols + h * 16];
#pragma unroll
    for (int q = 0; q < 4; ++q) {
        const float4 hv = hs[q];
        atomicAdd(&ad[q * 4 + 0], hv.x * alpha);
        atomicAdd(&ad[q * 4 + 1], hv.y * alpha);
        atomicAdd(&ad[q * 4 + 2], hv.z * alpha);
        atomicAdd(&ad[q * 4 + 3], hv.w * alpha);
    }
}

// Xout = elu(agg + b)
__global__ void k_finalize(const float* __restrict__ agg, const float* __restrict__ b,
                           float* __restrict__ Xout, int n, int ncols) {
    const int gid = blockIdx.x * blockDim.x + threadIdx.x;
    if (gid >= n * ncols) return;
    const int c = gid % ncols;
    const float v = agg[gid] + b[c];
    Xout[gid] = v > 0.f ? v : (__expf(v) - 1.0f);
}

// out[n] = sigmoid(X @ Wout + bout); embeddings appended after out
__global__ void k_output(const float* __restrict__ X, const float* __restrict__ Wout,
                         const float* __restrict__ bout, float* __restrict__ out, int n) {
    const int i = blockIdx.x * blockDim.x + threadIdx.x;
    if (i >= n) return;
    float acc = bout[0];
#pragma unroll
    for (int c = 0; c < 16; ++c) acc += X[i * 16 + c] * Wout[c];
    out[i] = 1.0f / (1.0f + __expf(-acc));
    float* emb = out + n;
    float4* ev = (float4*)&emb[i * 16];
    const float4* xv = (const float4*)&X[i * 16];
#pragma unroll
    for (int q = 0; q < 4; ++q) ev[q] = xv[q];
}

extern "C" void kernel_launch(void* const* d_in, const int* in_sizes, int n_in,
                              void* d_out, int out_size, void* d_ws, size_t ws_size,
                              hipStream_t stream) {
    const float* x   = (const float*)d_in[0];
    const int*   ei  = (const int*)d_in[1];
    const float* W1  = (const float*)d_in[2];
    const float* as1 = (const float*)d_in[3];
    const float* ad1 = (const float*)d_in[4];
    const float* b1  = (const float*)d_in[5];
    const float* W2  = (const float*)d_in[6];
    const float* as2 = (const float*)d_in[7];
    const float* ad2 = (const float*)d_in[8];
    const float* b2  = (const float*)d_in[9];
    const float* W3  = (const float*)d_in[10];
    const float* as3 = (const float*)d_in[11];
    const float* ad3 = (const float*)d_in[12];
    const float* b3  = (const float*)d_in[13];
    const float* Wo  = (const float*)d_in[14];
    const float* bo  = (const float*)d_in[15];

    const int N = N_NODES, E = N_EDGES, Etot = N_EDGES + N_NODES;

    float* ws   = (float*)d_ws;
    float* H    = ws;                              // N*32
    float* AGG  = ws + (size_t)N * 32;             // N*32
    float* XA   = ws + (size_t)N * 64;             // N*32
    float* XB   = ws + (size_t)N * 96;             // N*32
    float* ASRC = ws + (size_t)N * 128;            // N*2
    float* ADST = ASRC + (size_t)N * 2;            // N*2
    unsigned* MU = (unsigned*)(ADST + (size_t)N * 2); // N*2
    float* S    = (float*)(MU + (size_t)N * 2);    // N*2

    const int tiles  = (N + 15) / 16;
    const int pblks  = (tiles + 3) / 4;            // 4 waves (128 thr) per block

    auto attn = [&](const float* as, const float* ad, const float* b,
                    int heads, int ncols, float* xout) {
        k_alpha<<<(N + 255) / 256, 256, 0, stream>>>(H, as, ad, ASRC, ADST, MU, S, AGG,
                                                     N, heads, ncols);
        k_edge_max<<<(Etot + 255) / 256, 256, 0, stream>>>(ei, ASRC, ADST, MU, E, Etot, heads);
        k_edge_sum<<<(Etot + 255) / 256, 256, 0, stream>>>(ei, ASRC, ADST, MU, S, E, Etot, heads);
        k_edge_agg<<<(Etot * heads + 255) / 256, 256, 0, stream>>>(ei, ASRC, ADST, MU, S, H,
                                                                   AGG, E, Etot, heads, ncols);
        k_finalize<<<(N * ncols + 255) / 256, 256, 0, stream>>>(AGG, b, xout, N, ncols);
    };

    // layer 1: in=3, heads=2, out 32
    k_proj_t<3, 32><<<pblks, 128, 0, stream>>>(x, W1, H, N);
    attn(as1, ad1, b1, 2, 32, XA);
    // layer 2: in=32, heads=2, out 32
    k_proj_t<32, 32><<<pblks, 128, 0, stream>>>(XA, W2, H, N);
    attn(as2, ad2, b2, 2, 32, XB);
    // layer 3: in=32, heads=1, out 16
    k_proj_t<32, 16><<<pblks, 128, 0, stream>>>(XB, W3, H, N);
    attn(as3, ad3, b3, 1, 16, XA);                 // XA now holds [N,16] embeddings

    k_output<<<(N + 255) / 256, 256, 0, stream>>>(XA, Wo, bo, (float*)d_out, N);
}